// My_loss_8254927143083
// MI455X (gfx1250) — compile-verified
//
#include <hip/hip_runtime.h>
#include <math.h>

typedef __attribute__((ext_vector_type(2))) float v2f;
typedef __attribute__((ext_vector_type(4))) float v4f;
typedef __attribute__((ext_vector_type(8))) float v8f;

#define NUM_BATCHES      20
#define ELEMS_PER_BATCH  (128*128*128)          /* 2,097,152 */
#define BLOCKS_PER_BATCH 128
#define ELEMS_PER_BLOCK  (ELEMS_PER_BATCH / BLOCKS_PER_BATCH)  /* 16384 */
#define THREADS          256
#define ITERS            (ELEMS_PER_BLOCK / (THREADS * 4))     /* 16 */
#define NUM_BLOCKS       (NUM_BATCHES * BLOCKS_PER_BATCH)      /* 2560 */
#define N_TOTAL_D        ((double)NUM_BATCHES * (double)ELEMS_PER_BATCH)

// clip(ln v, -100) == ln2 * clip(log2 v, -100/ln2)
#define LOG2_CLAMP  (-144.26950408889634f)   /* -100 / ln(2) */
#define LN2_D       (0.6931471805599453)

// ---------------------------------------------------------------------------
// Sum 256 f32 values in LDS with 4 chained V_WMMA_F32_16X16X4_F32.
// B = ones(4x16) => D[m][n] = sum_k A[m][k] + C[m][n] (row sums). After the
// chain, per-lane sum of the 8 accumulator VGPRs + one lane-16 swap-add gives
// the full 256-element total in every lane of wave 0. Needs EXEC all ones.
// ---------------------------------------------------------------------------
__device__ __forceinline__ float wmma_reduce256(const float* __restrict__ lds,
                                                int lane) {
  v8f acc = {};
  v2f bones; bones.x = 1.0f; bones.y = 1.0f;
#pragma unroll
  for (int t = 0; t < 4; ++t) {
    v2f a;
    a.x = lds[t * 64 + lane];
    a.y = lds[t * 64 + 32 + lane];
    acc = __builtin_amdgcn_wmma_f32_16x16x4_f32(
        false, a, false, bones, (short)0, acc, false, false);
  }
  float s = acc[0] + acc[1] + acc[2] + acc[3] + acc[4] + acc[5] + acc[6] + acc[7];
  s += __shfl_xor(s, 16, 32);
  return s;
}

// ---------------------------------------------------------------------------
// Pass 1: streaming BCE (log2 domain) + wave-ballot predicate counts.
// ws layout: [0,2560) bce2 | [2560,5120) cnt_pred | [5120,7680) cnt_true
// ---------------------------------------------------------------------------
__global__ __launch_bounds__(THREADS)
void bce_pore_pass1(const float* __restrict__ x, const float* __restrict__ y,
                    float* __restrict__ part) {
  __shared__ float lds[THREADS];
  __shared__ int   wcp[THREADS / 32];
  __shared__ int   wct[THREADS / 32];

  const int tid  = threadIdx.x;
  const int wave = tid >> 5;
  const int lane = tid & 31;
  const int b    = blockIdx.x / BLOCKS_PER_BATCH;
  const int j    = blockIdx.x % BLOCKS_PER_BATCH;
  const long long base =
      (long long)b * ELEMS_PER_BATCH + (long long)j * ELEMS_PER_BLOCK;

  float bce2 = 0.0f;      // BCE partial in log2 domain (x ln2 deferred)
  int   icp  = 0;         // wave-uniform count of (x <= 0.5)
  int   ict  = 0;         // wave-uniform count of ((int)y == 0)

  for (int i = 0; i < ITERS; ++i) {
    const long long idx = base + (long long)i * (THREADS * 4) + tid * 4;
    v4f xv = __builtin_nontemporal_load((const v4f*)(x + idx));
    v4f yv = __builtin_nontemporal_load((const v4f*)(y + idx));
    if (i + 1 < ITERS) {
      __builtin_prefetch(x + idx + THREADS * 4, 0, 1);
      __builtin_prefetch(y + idx + THREADS * 4, 0, 1);
    }
#pragma unroll
    for (int k = 0; k < 4; ++k) {
      const float xx = xv[k];
      const float yy = yv[k];
      // hardware v_log_f32; 1-x exact for x>=0.5 (Sterbenz), ~1e-7 rel below
      const float l2x  = fmaxf(__log2f(xx),        LOG2_CLAMP);
      const float l2mx = fmaxf(__log2f(1.0f - xx), LOG2_CLAMP);
      bce2 += fmaf(yy, l2x - l2mx, l2mx);   // y*l2x + (1-y)*l2mx
#if __has_builtin(__builtin_amdgcn_ballot_w32)
      icp += __builtin_popcount(__builtin_amdgcn_ballot_w32(xx <= 0.5f));
      ict += __builtin_popcount(__builtin_amdgcn_ballot_w32((int)yy == 0));
#else
      icp += (xx <= 0.5f) ? 32 : 0;   // unreachable fallback keeps types sane
      ict += ((int)yy == 0) ? 32 : 0;
#endif
    }
  }

#if !__has_builtin(__builtin_amdgcn_ballot_w32)
  // Fallback: per-lane counts -> wave sums via shfl (not expected to compile in)
  for (int m = 16; m >= 1; m >>= 1) {
    icp += __shfl_xor(icp, m, 32);
    ict += __shfl_xor(ict, m, 32);
  }
  icp /= 32; ict /= 32;  // undo the *32 scaling trick
#endif

  if (lane == 0) { wcp[wave] = icp; wct[wave] = ict; }
  lds[tid] = bce2;
  __syncthreads();

  if (tid < 32) {
    const float tb = wmma_reduce256(lds, tid);
    if (tid == 0) {
      int scp = 0, sct = 0;
#pragma unroll
      for (int q = 0; q < THREADS / 32; ++q) { scp += wcp[q]; sct += wct[q]; }
      part[blockIdx.x]                  = tb;
      part[NUM_BLOCKS + blockIdx.x]     = (float)scp;
      part[2 * NUM_BLOCKS + blockIdx.x] = (float)sct;
    }
  }
}

// ---------------------------------------------------------------------------
// Pass 2: one 640-thread block (20 waves). Wave w reduces batch w's 128 count
// partials; all threads fold the 2560 BCE partials in f64; thread 0 combines.
// ---------------------------------------------------------------------------
__global__ __launch_bounds__(640)
void bce_pore_pass2(const float* __restrict__ part, float* __restrict__ out) {
  __shared__ double dsum[NUM_BATCHES];
  __shared__ float  poresq[NUM_BATCHES];

  const int tid  = threadIdx.x;      // 0..639
  const int w    = tid >> 5;         // wave == batch id
  const int lane = tid & 31;

  // Per-batch predicate counts: 128 block partials per batch, 4 per lane.
  const int cbase = NUM_BLOCKS + w * BLOCKS_PER_BATCH;
  float cp = 0.0f, ct = 0.0f;
#pragma unroll
  for (int q = 0; q < BLOCKS_PER_BATCH / 32; ++q) {
    cp += part[cbase + q * 32 + lane];
    ct += part[NUM_BLOCKS + cbase + q * 32 + lane];
  }
#pragma unroll
  for (int m = 16; m >= 1; m >>= 1) {
    cp += __shfl_xor(cp, m, 32);
    ct += __shfl_xor(ct, m, 32);
  }
  if (lane == 0) {
    const float inv = 1.0f / (float)ELEMS_PER_BATCH;
    const float d = cp * inv - ct * inv;   // por_pred - por_true
    poresq[w] = d * d;
  }

  // BCE grand total in f64 (2560 partials, 4 per thread).
  double acc = 0.0;
#pragma unroll
  for (int q = 0; q < NUM_BLOCKS / 640; ++q) acc += (double)part[tid + q * 640];
#pragma unroll
  for (int m = 16; m >= 1; m >>= 1) acc += __shfl_xor(acc, m, 32);
  if (lane == 0) dsum[w] = acc;
  __syncthreads();

  if (tid == 0) {
    double tot = 0.0;
    for (int i = 0; i < NUM_BATCHES; ++i) tot += dsum[i];
    float pore = 0.0f;
    for (int i = 0; i < NUM_BATCHES; ++i) pore += poresq[i];
    pore *= (1.0f / (float)NUM_BATCHES);
    // deferred ln2 conversion from log2 domain, then mean + negate
    const float mse = (float)(-(tot * LN2_D) / N_TOTAL_D);
    out[0] = pore + mse;   // PORE + mseloss
    out[1] = pore;         // PORE
  }
}

extern "C" void kernel_launch(void* const* d_in, const int* in_sizes, int n_in,
                              void* d_out, int out_size, void* d_ws, size_t ws_size,
                              hipStream_t stream) {
  const float* x = (const float*)d_in[0];
  const float* y = (const float*)d_in[1];
  float* part = (float*)d_ws;          // 3 * 2560 * 4 B = 30 KB scratch
  float* out  = (float*)d_out;

  bce_pore_pass1<<<NUM_BLOCKS, THREADS, 0, stream>>>(x, y, part);
  bce_pore_pass2<<<1, 640, 0, stream>>>(part, out);
}